// Composer_81707457839189
// MI455X (gfx1250) — compile-verified
//
#include <hip/hip_runtime.h>
#include <math.h>

typedef __attribute__((ext_vector_type(2))) float v2f;
typedef __attribute__((ext_vector_type(8))) float v8f;

#define EULER_F 2.71828182845904523536f
#define LOG2E_F 1.44269504088896340736f
#define LN2_F   0.69314718055994530942f

// order-preserving float <-> uint mapping so we can use integer atomicMin
__device__ __forceinline__ unsigned ordKey(float f) {
    unsigned u = __float_as_uint(f);
    return (u & 0x80000000u) ? ~u : (u ^ 0x80000000u);
}
__device__ __forceinline__ float ordInv(unsigned k) {
    unsigned u = (k & 0x80000000u) ? (k ^ 0x80000000u) : ~k;
    return __uint_as_float(u);
}

__global__ void init_min_kernel(unsigned* __restrict__ mins) {
    if (threadIdx.x < 16) mins[threadIdx.x] = 0xFFFFFFFFu;
}

// per-column min over (N,16) row-major, float4 loads (column set fixed per thread)
__global__ void colmin_kernel(const float* __restrict__ V,
                              unsigned* __restrict__ mins, int n4) {
    __shared__ unsigned smin[16];
    if (threadIdx.x < 16) smin[threadIdx.x] = 0xFFFFFFFFu;
    __syncthreads();
    const int gt = blockIdx.x * blockDim.x + threadIdx.x;
    const int stride = gridDim.x * blockDim.x;   // *4 elements => multiple of 16
    float m0 = INFINITY, m1 = INFINITY, m2 = INFINITY, m3 = INFINITY;
    const float4* V4 = (const float4*)V;
    for (int t = gt; t < n4; t += stride) {
        float4 v = V4[t];
        m0 = fminf(m0, v.x); m1 = fminf(m1, v.y);
        m2 = fminf(m2, v.z); m3 = fminf(m3, v.w);
    }
    const int c0 = (4 * gt) & 15;                // 0,4,8,12 — fixed per thread
    atomicMin(&smin[c0 + 0], ordKey(m0));
    atomicMin(&smin[c0 + 1], ordKey(m1));
    atomicMin(&smin[c0 + 2], ordKey(m2));
    atomicMin(&smin[c0 + 3], ordKey(m3));
    __syncthreads();
    if (threadIdx.x < 16) atomicMin(&mins[threadIdx.x], smin[threadIdx.x]);
}

// One row (16 modes) per wave32. modal = P * ones via chained v_wmma_f32_16x16x4_f32.
// All transcendentals via raw v_log_f32 / v_exp_f32 (base-2): pow(x,y) = exp2(y*log2(x)).
__global__ void __launch_bounds__(256)
energy_kernel(const float* __restrict__ V,
              const float* __restrict__ w,
              const float* __restrict__ b,
              const float* __restrict__ gammas,
              const unsigned* __restrict__ mins,
              float* __restrict__ alphas,
              float* __restrict__ logits,
              int N) {
    const int lane = threadIdx.x & 31;
    const int wid  = (blockIdx.x * blockDim.x + threadIdx.x) >> 5;
    const int nw   = (gridDim.x * blockDim.x) >> 5;
    const int i    = lane & 15;            // this lane's mode (A-matrix M index)
    const int hi   = (lane >> 4) << 1;     // K offset within chunk: 0 or 2

    const float tau   = 0.25f;             // 1/sqrt(16)
    const float vminI = ordInv(mins[i]);
    const float wi = w[i], bi = b[i];

    // Per-lane constants for its 8 (i,j) slots: j index, gamma, diagonal flag
    int   jj[8];
    float g[8];
    bool  dg[8];
#pragma unroll
    for (int s = 0; s < 8; ++s) {
        const int j = 4 * (s >> 1) + hi + (s & 1);
        jj[s] = j;
        dg[s] = (i == j);
        float gv;
        if (i == j)      gv = 0.0f;
        else if (i < j)  gv = gammas[i * 16 + j];
        else             gv = gammas[j * 16 + i];
        g[s] = gv;
    }

    v2f bones; bones[0] = 1.0f; bones[1] = 1.0f;   // B = ones(4,16) => D = row-sums of A

    for (int n = wid; n < N; n += nw) {
        const float vv  = V[n * 16 + i];
        const float vc  = fmaxf(vv - vminI + EULER_F, EULER_F);
        const float phi = (1.0f + wi) * vc + bi;           // > 0 by construction
        const float l2p = __builtin_amdgcn_logf(phi);      // log2(phi_i)

        // batch all 8 cross-lane gathers first (one wait, overlapped latency)
        float l2j[8];
#pragma unroll
        for (int s = 0; s < 8; ++s) l2j[s] = __shfl(l2p, jj[s], 32);

        // P[i,j] = phi_i^g * phi_j^(1-g) = exp2(l2j + g*(l2p - l2j)); diag = phi_i
        float pv[8];
#pragma unroll
        for (int s = 0; s < 8; ++s) {
            const float e = fmaf(g[s], l2p - l2j[s], l2j[s]);
            const float p = __builtin_amdgcn_exp2f(e);
            pv[s] = dg[s] ? phi : p;
        }

        // modal = row-sums of P via 4 chained WMMAs over K-chunks of 4
        v8f acc = {};
#pragma unroll
        for (int c = 0; c < 4; ++c) {
            v2f a; a[0] = pv[2 * c]; a[1] = pv[2 * c + 1];
            acc = __builtin_amdgcn_wmma_f32_16x16x4_f32(
                false, a, false, bones, (short)0, acc, false, false);
        }
        // lane l (<16): acc[v] = modal[v];  lane l (>=16): acc[v] = modal[v+8]

        // z_k = -tau * modal_k ; softmax stats over 8 local + cross-half combine
        float z[8];
#pragma unroll
        for (int k = 0; k < 8; ++k) z[k] = -tau * acc[k];
        float mloc = z[0];
#pragma unroll
        for (int k = 1; k < 8; ++k) mloc = fmaxf(mloc, z[k]);
        const float mall = fmaxf(mloc, __shfl_xor(mloc, 16, 32));

        const float mb = -mall * LOG2E_F;
        float sloc = 0.0f;
#pragma unroll
        for (int k = 0; k < 8; ++k)
            sloc += __builtin_amdgcn_exp2f(fmaf(z[k], LOG2E_F, mb));
        const float sall = sloc + __shfl_xor(sloc, 16, 32);
        const float logZ = fmaf(__builtin_amdgcn_logf(sall), LN2_F, mall);

        // x = z[lane & 7] via branchless static select chain
        const int li = lane & 7;
        float x = z[0];
        x = (li == 1) ? z[1] : x;
        x = (li == 2) ? z[2] : x;
        x = (li == 3) ? z[3] : x;
        x = (li == 4) ? z[4] : x;
        x = (li == 5) ? z[5] : x;
        x = (li == 6) ? z[6] : x;
        x = (li == 7) ? z[7] : x;

        const float lg = x - logZ;                              // logit
        const float al = __builtin_amdgcn_exp2f(lg * LOG2E_F);  // alpha = exp(logit)
        const int col  = (lane & 7) + ((lane >> 4) << 3);
        const long base = (long)n * 16 + col;
        if ((lane & 8) == 0) logits[base] = lg;  // lanes 0-7,16-23
        else                 alphas[base] = al;  // lanes 8-15,24-31
    }
}

extern "C" void kernel_launch(void* const* d_in, const int* in_sizes, int n_in,
                              void* d_out, int out_size, void* d_ws, size_t ws_size,
                              hipStream_t stream) {
    (void)n_in; (void)out_size; (void)ws_size;
    const float* V      = (const float*)d_in[0];
    const float* w      = (const float*)d_in[1];
    const float* b      = (const float*)d_in[2];
    const float* gammas = (const float*)d_in[3];

    const int NM = in_sizes[0];
    const int N  = NM / 16;

    unsigned* mins = (unsigned*)d_ws;
    float* alphas  = (float*)d_out;
    float* logits  = alphas + (size_t)N * 16;

    init_min_kernel<<<1, 16, 0, stream>>>(mins);
    colmin_kernel<<<1024, 256, 0, stream>>>(V, mins, NM / 4);
    energy_kernel<<<2048, 256, 0, stream>>>(V, w, b, gammas, mins, alphas, logits, N);
}